// A12_TFN_59811714564651
// MI455X (gfx1250) — compile-verified
//
#include <hip/hip_runtime.h>
#include <hip/hip_bf16.h>
#include <math.h>

// ---------------------------------------------------------------------------
// Types for WMMA f32 16x16x4 (CDNA5 / gfx1250, wave32)
// A: 16x4 f32  -> 2 VGPRs/lane (lane = M row; half-wave selects K {0,1} vs {2,3})
// B: 4x16 f32  -> 2 VGPRs/lane (lane = N col; half-wave selects K {0,1} vs {2,3})
// C/D: 16x16 f32 -> 8 VGPRs/lane (elem r: M = r + 8*(lane>=16), N = lane&15)
// ---------------------------------------------------------------------------
typedef __attribute__((ext_vector_type(2))) float v2f;
typedef __attribute__((ext_vector_type(8))) float v8f;

__device__ __forceinline__ v8f wmma4(v2f a, v2f b, v8f c) {
  return __builtin_amdgcn_wmma_f32_16x16x4_f32(
      /*neg_a=*/false, a, /*neg_b=*/false, b,
      /*c_mod=*/(short)0, c, /*reuse_a=*/false, /*reuse_b=*/false);
}

__device__ __forceinline__ float sigmoidf_(float x) {
  return 1.0f / (1.0f + __expf(-x));
}

// async global->LDS (gfx1250 ASYNCcnt path), guarded for toolchain drift
#if defined(__has_builtin)
#if __has_builtin(__builtin_amdgcn_global_load_async_to_lds_b32) && \
    __has_builtin(__builtin_amdgcn_s_wait_asynccnt)
#define TFN_ASYNC 1
#endif
#endif

// builtin signature (per hipcc diagnostic): (as1 int*, as3 int*, imm, imm)
typedef __attribute__((address_space(1))) int* tfn_gas_ip;
typedef __attribute__((address_space(3))) int* tfn_las_ip;

// ---------------------------------------------------------------------------
// Problem constants
// ---------------------------------------------------------------------------
#define BATCH   128
#define SEQ     50
#define TIN     300
#define AIN     74
#define VIN     35
#define HID     32        // audio/video hidden
#define TH      128       // text hidden / text out
#define G4      512       // 4*TH (lstm gates)
#define PF      768
#define KFUS    140481    // (128+1)*(32+1)*(32+1)
#define AV      1089      // 33*33
#define NSPLIT  8
#define KCHUNK  17564     // multiple of 4; 8*17564 >= KFUS
#define KTILE   256       // k-extent staged to LDS per round
#define KSTEPS  (KTILE / 4)
#define LROW    260       // LDS row stride in floats (== 4 mod 64 banks)

// ---------------------------------------------------------------------------
// 1. audio/video subnet:  y = relu(relu(relu((x*g+b)W1^T+b1)W2^T+b2)W3^T+b3)
// ---------------------------------------------------------------------------
__global__ void tfn_subnet_kernel(const float* __restrict__ x,
                                  const float* __restrict__ gamma,
                                  const float* __restrict__ beta,
                                  const float* __restrict__ w1, const float* __restrict__ b1,
                                  const float* __restrict__ w2, const float* __restrict__ b2,
                                  const float* __restrict__ w3, const float* __restrict__ b3,
                                  float* __restrict__ out_aug, int in_dim) {
  __shared__ float xs[AIN];
  __shared__ float y0[HID];
  __shared__ float y1[HID];
  const int b = blockIdx.x;
  const int h = threadIdx.x;  // 0..31

  for (int k = h; k < in_dim; k += 32)
    xs[k] = x[b * in_dim + k] * gamma[k] + beta[k];
  __syncthreads();

  float acc = b1[h];
  for (int k = 0; k < in_dim; ++k) acc += xs[k] * w1[h * in_dim + k];
  y0[h] = fmaxf(acc, 0.0f);
  __syncthreads();

  acc = b2[h];
  for (int k = 0; k < HID; ++k) acc += y0[k] * w2[h * HID + k];
  y1[h] = fmaxf(acc, 0.0f);
  __syncthreads();

  acc = b3[h];
  for (int k = 0; k < HID; ++k) acc += y1[k] * w3[h * HID + k];
  if (h == 0) out_aug[b * 33] = 1.0f;
  out_aug[b * 33 + 1 + h] = fmaxf(acc, 0.0f);
}

// ---------------------------------------------------------------------------
// 2. av outer product: av[b, a*33+v] = a_aug[b,a] * v_aug[b,v]
// ---------------------------------------------------------------------------
__global__ void tfn_av_kernel(const float* __restrict__ a_aug,
                              const float* __restrict__ v_aug,
                              float* __restrict__ av) {
  int i = blockIdx.x * blockDim.x + threadIdx.x;
  if (i >= BATCH * AV) return;
  int b = i / AV;
  int j = i - b * AV;
  av[i] = a_aug[b * 33 + j / 33] * v_aug[b * 33 + j % 33];
}

// ---------------------------------------------------------------------------
// 3. init helpers
// ---------------------------------------------------------------------------
__global__ void tfn_zero_kernel(float* __restrict__ p, int n) {
  int i = blockIdx.x * blockDim.x + threadIdx.x;
  if (i < n) p[i] = 0.0f;
}
__global__ void tfn_tones_kernel(float* __restrict__ t_aug) {
  int b = threadIdx.x;
  if (b < BATCH) t_aug[b * 129] = 1.0f;
}

// ---------------------------------------------------------------------------
// 4. x_proj = text_x @ w_ih^T + (b_ih + b_hh)   [M=B*S=6400, N=512, K=300]
// ---------------------------------------------------------------------------
__global__ void tfn_xproj_kernel(const float* __restrict__ x,
                                 const float* __restrict__ w_ih,
                                 const float* __restrict__ b_ih,
                                 const float* __restrict__ b_hh,
                                 float* __restrict__ xp) {
  const int lane = threadIdx.x & 31;
  const int wave = threadIdx.x >> 5;
  const int tile = blockIdx.x * 4 + wave;       // 400*32 = 12800 tiles
  const int mt = tile >> 5;
  const int nt = tile & 31;
  const int mA = mt * 16 + (lane & 15);
  const int nB = nt * 16 + (lane & 15);
  const int kh = (lane >> 4) << 1;

  const float* arow = x + (long)mA * TIN;
  const float* brow = w_ih + (long)nB * TIN;
  v8f acc = {};
  for (int kb = 0; kb < TIN; kb += 4) {
    v2f a = *(const v2f*)(arow + kb + kh);
    v2f b = *(const v2f*)(brow + kb + kh);
    acc = wmma4(a, b, acc);
  }
  const float bias = b_ih[nB] + b_hh[nB];
  const int mbase = mt * 16 + ((lane >> 4) << 3);
  const int nC = nt * 16 + (lane & 15);
  for (int r = 0; r < 8; ++r)
    xp[(long)(mbase + r) * G4 + nC] = acc[r] + bias;
}

// ---------------------------------------------------------------------------
// 5. one LSTM step: gates = xp[:,t,:] + h_in @ w_hh^T ; update c, h_out.
// ---------------------------------------------------------------------------
__global__ void tfn_lstm_step_kernel(const float* __restrict__ xp,
                                     const float* __restrict__ w_hh,
                                     const float* __restrict__ h_in,
                                     float* __restrict__ h_out,
                                     float* __restrict__ c,
                                     int t) {
  const int lane = threadIdx.x & 31;
  const int bt = blockIdx.x >> 3;
  const int ht = blockIdx.x & 7;
  const int mA = bt * 16 + (lane & 15);
  const int kh = (lane >> 4) << 1;
  const int nloc = ht * 16 + (lane & 15);

  const float* ar = h_in + mA * TH;
  const float* bi = w_hh + (long)(0 * TH + nloc) * TH;
  const float* bf = w_hh + (long)(1 * TH + nloc) * TH;
  const float* bg = w_hh + (long)(2 * TH + nloc) * TH;
  const float* bo = w_hh + (long)(3 * TH + nloc) * TH;

  v8f ci = {}, cf = {}, cg = {}, co = {};
  for (int kb = 0; kb < TH; kb += 4) {
    v2f a = *(const v2f*)(ar + kb + kh);
    ci = wmma4(a, *(const v2f*)(bi + kb + kh), ci);
    cf = wmma4(a, *(const v2f*)(bf + kb + kh), cf);
    cg = wmma4(a, *(const v2f*)(bg + kb + kh), cg);
    co = wmma4(a, *(const v2f*)(bo + kb + kh), co);
  }

  const int mbase = bt * 16 + ((lane >> 4) << 3);
  for (int r = 0; r < 8; ++r) {
    const int m = mbase + r;
    const float* xpr = xp + ((long)m * SEQ + t) * G4;
    const float gi = sigmoidf_(ci[r] + xpr[0 * TH + nloc]);
    const float gf = sigmoidf_(cf[r] + xpr[1 * TH + nloc]);
    const float gg = tanhf(cg[r] + xpr[2 * TH + nloc]);
    const float go = sigmoidf_(co[r] + xpr[3 * TH + nloc]);
    const float cn = gf * c[m * TH + nloc] + gi * gg;
    c[m * TH + nloc] = cn;
    h_out[m * TH + nloc] = go * tanhf(cn);
  }
}

// ---------------------------------------------------------------------------
// 6. text projection: t_aug[:,1:] = hT @ t_w^T + t_b   [128,128,K=128]
// ---------------------------------------------------------------------------
__global__ void tfn_tproj_kernel(const float* __restrict__ h,
                                 const float* __restrict__ t_w,
                                 const float* __restrict__ t_b,
                                 float* __restrict__ t_aug) {
  const int lane = threadIdx.x & 31;
  const int mt = blockIdx.x >> 3;
  const int nt = blockIdx.x & 7;
  const int mA = mt * 16 + (lane & 15);
  const int nB = nt * 16 + (lane & 15);
  const int kh = (lane >> 4) << 1;

  const float* ar = h + mA * TH;
  const float* br = t_w + nB * TH;
  v8f acc = {};
  for (int kb = 0; kb < TH; kb += 4) {
    v2f a = *(const v2f*)(ar + kb + kh);
    v2f b = *(const v2f*)(br + kb + kh);
    acc = wmma4(a, b, acc);
  }
  const float bias = t_b[nB];
  const int mbase = mt * 16 + ((lane >> 4) << 3);
  const int nC = nt * 16 + (lane & 15);
  for (int r = 0; r < 8; ++r)
    t_aug[(mbase + r) * 129 + 1 + nC] = acc[r] + bias;
}

// ---------------------------------------------------------------------------
// 7. BIG GEMM: y1_partial = fusion @ pf_w1^T, fusion generated on the fly
//    from fusion[m,k] = t_aug[m, k/1089] * av[m, k%1089] (incremental T/j).
//    W1 tiles (16 rows x 256 k) are staged into LDS via async global->LDS,
//    double-buffered; all 8 waves (8 batch tiles) consume the same LDS tile
//    so each weight byte is fetched from L2/HBM exactly once per workgroup.
// ---------------------------------------------------------------------------
__global__ __launch_bounds__(256)
void tfn_pf1_kernel(const float* __restrict__ t_aug,
                    const float* __restrict__ av,
                    const float* __restrict__ w1,
                    float* __restrict__ partial) {
  __shared__ float lds[2][16 * LROW];

  const int tid = threadIdx.x;
  const int lane = tid & 31;
  const int wt = tid >> 5;                 // batch tile 0..7
  const int nt = blockIdx.x;               // 0..47
  const int ks = blockIdx.y;               // 0..7

  const int k0 = ks * KCHUNK;
  int k1 = k0 + KCHUNK;
  if (k1 > KFUS) k1 = KFUS;
  const int ntiles = (KCHUNK + KTILE - 1) / KTILE;   // 69

  // staging role: thread -> (row srow, 16 consecutive cols at scol)
  const int srow = tid >> 4;               // 0..15
  const int scol = (tid & 15) << 4;        // 0..240
  const float* wsrc = w1 + (size_t)(nt * 16 + srow) * KFUS;

  // compute-side fragment setup
  const int mA = wt * 16 + (lane & 15);
  const int kh = (lane >> 4) << 1;
  const float* trow = t_aug + mA * 129;
  const float* avrow = av + mA * AV;

  // incremental (T, j) for this lane's element-0 k index
  int kk0 = k0 + kh;
  int T = kk0 / AV;
  int j = kk0 - T * AV;

  // ---- stage tile 0 ----
  {
    float* dst = &lds[0][srow * LROW + scol];
    const int gk = k0 + scol;
#if TFN_ASYNC
    for (int e = 0; e < 16; ++e) {
      if (gk + e < k1)
        __builtin_amdgcn_global_load_async_to_lds_b32(
            (tfn_gas_ip)(wsrc + gk + e), (tfn_las_ip)(dst + e), 0, 0);
      else
        dst[e] = 0.0f;
    }
    __builtin_amdgcn_s_wait_asynccnt(0);
#else
    for (int e = 0; e < 16; ++e)
      dst[e] = (gk + e < k1) ? wsrc[gk + e] : 0.0f;
#endif
  }
  __syncthreads();

  v8f acc = {};
  for (int tile = 0; tile < ntiles; ++tile) {
    const int buf = tile & 1;

    // ---- issue async stage of next tile into the other buffer ----
    if (tile + 1 < ntiles) {
      float* dst = &lds[buf ^ 1][srow * LROW + scol];
      const int gk = k0 + (tile + 1) * KTILE + scol;
#if TFN_ASYNC
      for (int e = 0; e < 16; ++e) {
        if (gk + e < k1)
          __builtin_amdgcn_global_load_async_to_lds_b32(
              (tfn_gas_ip)(wsrc + gk + e), (tfn_las_ip)(dst + e), 0, 0);
        else
          dst[e] = 0.0f;
      }
#else
      for (int e = 0; e < 16; ++e)
        dst[e] = (gk + e < k1) ? wsrc[gk + e] : 0.0f;
#endif
    }

    // ---- consume current buffer: 64 WMMA k-steps ----
    const float* bbase = &lds[buf][(lane & 15) * LROW + kh];
    for (int s = 0; s < KSTEPS; ++s) {
      int T1 = T, j1 = j + 1;
      if (j1 == AV) { j1 = 0; T1 = T + 1; }
      const int Tc0 = (T <= 128) ? T : 0;    // clamp for zero-padded tail
      const int Tc1 = (T1 <= 128) ? T1 : 0;
      v2f a, b;
      a.x = trow[Tc0] * avrow[j];
      a.y = trow[Tc1] * avrow[j1];
      b = *(const v2f*)(bbase + s * 4);      // bank-conflict-free (stride 260)
      acc = wmma4(a, b, acc);
      j += 4;
      if (j >= AV) { j -= AV; T += 1; }
    }

#if TFN_ASYNC
    __builtin_amdgcn_s_wait_asynccnt(0);
#endif
    __syncthreads();
  }

  float* pout = partial + (long)ks * (BATCH * PF);
  const int mbase = wt * 16 + ((lane >> 4) << 3);
  const int nC = nt * 16 + (lane & 15);
  for (int r = 0; r < 8; ++r)
    pout[(mbase + r) * PF + nC] = acc[r];
}

// 7b. deterministic split-K reduction + bias + relu
__global__ void tfn_pf1_reduce_kernel(const float* __restrict__ partial,
                                      const float* __restrict__ b1,
                                      float* __restrict__ y1) {
  int i = blockIdx.x * blockDim.x + threadIdx.x;
  if (i >= BATCH * PF) return;
  const int n = i % PF;
  float s = b1[n];
  for (int k = 0; k < NSPLIT; ++k) s += partial[(long)k * (BATCH * PF) + i];
  y1[i] = fmaxf(s, 0.0f);
}

// ---------------------------------------------------------------------------
// 8. y2 = relu(y1 @ pf_w2^T + b2)  [128,768,K=768]
// ---------------------------------------------------------------------------
__global__ void tfn_pf2_kernel(const float* __restrict__ y1,
                               const float* __restrict__ w2,
                               const float* __restrict__ b2,
                               float* __restrict__ y2) {
  const int lane = threadIdx.x & 31;
  const int mt = blockIdx.x / 48;
  const int nt = blockIdx.x % 48;
  const int mA = mt * 16 + (lane & 15);
  const int nB = nt * 16 + (lane & 15);
  const int kh = (lane >> 4) << 1;

  const float* ar = y1 + (long)mA * PF;
  const float* br = w2 + (long)nB * PF;
  v8f acc = {};
  for (int kb = 0; kb < PF; kb += 4) {
    v2f a = *(const v2f*)(ar + kb + kh);
    v2f b = *(const v2f*)(br + kb + kh);
    acc = wmma4(a, b, acc);
  }
  const float bias = b2[nB];
  const int mbase = mt * 16 + ((lane >> 4) << 3);
  const int nC = nt * 16 + (lane & 15);
  for (int r = 0; r < 8; ++r)
    y2[(mbase + r) * PF + nC] = fmaxf(acc[r] + bias, 0.0f);
}

// ---------------------------------------------------------------------------
// 9. z = y2 @ pf_w3^T + b3; out = sigmoid(z)*6 - 3
// ---------------------------------------------------------------------------
__global__ void tfn_pf3_kernel(const float* __restrict__ y2,
                               const float* __restrict__ w3,
                               const float* __restrict__ b3,
                               float* __restrict__ out) {
  const int b = threadIdx.x;
  float s = b3[0];
  for (int p = 0; p < PF; ++p) s += y2[b * PF + p] * w3[p];
  out[b] = 6.0f * sigmoidf_(s) - 3.0f;
}

// ---------------------------------------------------------------------------
// launch
// ---------------------------------------------------------------------------
extern "C" void kernel_launch(void* const* d_in, const int* in_sizes, int n_in,
                              void* d_out, int out_size, void* d_ws, size_t ws_size,
                              hipStream_t stream) {
  const float* text_x  = (const float*)d_in[0];
  const float* audio_x = (const float*)d_in[1];
  const float* video_x = (const float*)d_in[2];
  const float* a_gamma = (const float*)d_in[3];
  const float* a_beta  = (const float*)d_in[4];
  const float* a_w1 = (const float*)d_in[5];  const float* a_b1 = (const float*)d_in[6];
  const float* a_w2 = (const float*)d_in[7];  const float* a_b2 = (const float*)d_in[8];
  const float* a_w3 = (const float*)d_in[9];  const float* a_b3 = (const float*)d_in[10];
  const float* v_gamma = (const float*)d_in[11];
  const float* v_beta  = (const float*)d_in[12];
  const float* v_w1 = (const float*)d_in[13]; const float* v_b1 = (const float*)d_in[14];
  const float* v_w2 = (const float*)d_in[15]; const float* v_b2 = (const float*)d_in[16];
  const float* v_w3 = (const float*)d_in[17]; const float* v_b3 = (const float*)d_in[18];
  const float* w_ih = (const float*)d_in[19]; const float* w_hh = (const float*)d_in[20];
  const float* b_ih = (const float*)d_in[21]; const float* b_hh = (const float*)d_in[22];
  const float* t_w  = (const float*)d_in[23]; const float* t_b  = (const float*)d_in[24];
  const float* pf_w1 = (const float*)d_in[25]; const float* pf_b1 = (const float*)d_in[26];
  const float* pf_w2 = (const float*)d_in[27]; const float* pf_b2 = (const float*)d_in[28];
  const float* pf_w3 = (const float*)d_in[29]; const float* pf_b3 = (const float*)d_in[30];

  // workspace layout (floats)
  float* ws = (float*)d_ws;
  float* a_aug   = ws;                                  // 128*33
  float* v_aug   = a_aug + BATCH * 33;                  // 128*33
  float* t_aug   = v_aug + BATCH * 33;                  // 128*129
  float* av      = t_aug + BATCH * 129;                 // 128*1089
  float* h0      = av + BATCH * AV;                     // 128*128
  float* cbuf    = h0 + BATCH * TH;                     // 128*128 (adjacent to h0)
  float* h1      = cbuf + BATCH * TH;                   // 128*128
  float* xp      = h1 + BATCH * TH;                     // 128*50*512
  float* y1      = xp + (long)BATCH * SEQ * G4;         // 128*768
  float* y2      = y1 + BATCH * PF;                     // 128*768
  float* partial = y2 + BATCH * PF;                     // 8*128*768

  // 1) audio / video subnets -> augmented rows
  tfn_subnet_kernel<<<BATCH, 32, 0, stream>>>(audio_x, a_gamma, a_beta,
      a_w1, a_b1, a_w2, a_b2, a_w3, a_b3, a_aug, AIN);
  tfn_subnet_kernel<<<BATCH, 32, 0, stream>>>(video_x, v_gamma, v_beta,
      v_w1, v_b1, v_w2, v_b2, v_w3, v_b3, v_aug, VIN);

  // 2) av outer product
  tfn_av_kernel<<<(BATCH * AV + 255) / 256, 256, 0, stream>>>(a_aug, v_aug, av);

  // 3) init h0, c = 0 (contiguous) and t_aug leading ones
  tfn_zero_kernel<<<(2 * BATCH * TH + 255) / 256, 256, 0, stream>>>(h0, 2 * BATCH * TH);
  tfn_tones_kernel<<<1, BATCH, 0, stream>>>(t_aug);

  // 4) x_proj (WMMA GEMM, 12800 tiles, 4 waves/block)
  tfn_xproj_kernel<<<3200, 128, 0, stream>>>(text_x, w_ih, b_ih, b_hh, xp);

  // 5) 50 sequential LSTM steps, ping-pong h buffers
  float* hbuf[2] = {h0, h1};
  for (int t = 0; t < SEQ; ++t) {
    tfn_lstm_step_kernel<<<64, 32, 0, stream>>>(
        xp, w_hh, hbuf[t & 1], hbuf[(t + 1) & 1], cbuf, t);
  }
  // final hidden state is in h0 (SEQ even)

  // 6) text projection into t_aug[:,1:]
  tfn_tproj_kernel<<<64, 32, 0, stream>>>(h0, t_w, t_b, t_aug);

  // 7) big fused-fusion GEMM, split-K deterministic, async-LDS staged
  tfn_pf1_kernel<<<dim3(48, NSPLIT), 256, 0, stream>>>(t_aug, av, pf_w1, partial);
  tfn_pf1_reduce_kernel<<<(BATCH * PF + 255) / 256, 256, 0, stream>>>(partial, pf_b1, y1);

  // 8) pf layer 2
  tfn_pf2_kernel<<<8 * 48, 32, 0, stream>>>(y1, pf_w2, pf_b2, y2);

  // 9) head
  tfn_pf3_kernel<<<1, BATCH, 0, stream>>>(y2, pf_w3, pf_b3, (float*)d_out);
}